// GraphSAGEEmbedder_42777874268288
// MI455X (gfx1250) — compile-verified
//
#include <hip/hip_runtime.h>
#include <hip/hip_bf16.h>

// ---------------------------------------------------------------------------
// GraphSAGE 3-layer forward for MI455X (gfx1250, wave32).
//   per layer: mean-scatter over edges -> dual GEMM (WMMA f32 16x16x4)
//              -> bias -> row L2 normalize -> (ReLU).
// ---------------------------------------------------------------------------

typedef __attribute__((ext_vector_type(2))) float v2f;
typedef __attribute__((ext_vector_type(8))) float v8f;

#define SAGE_EPS 1e-12f

// ---------------- zero workspace (agg + cnt) -------------------------------
__global__ __launch_bounds__(256) void sage_zero4(float4* __restrict__ p,
                                                  long long n4) {
  long long i = (long long)blockIdx.x * 256 + threadIdx.x;
  if (i < n4) p[i] = make_float4(0.f, 0.f, 0.f, 0.f);
}

// ---------------- edge gather + atomic mean-scatter ------------------------
template <int K>
__global__ __launch_bounds__(256) void sage_scatter(
    const float* __restrict__ xsrc, const int* __restrict__ src,
    const int* __restrict__ dst, float* __restrict__ agg,
    float* __restrict__ cnt, int E) {
  constexpr int C = K / 4;  // float4 chunks per feature row
  long long tid = (long long)blockIdx.x * 256 + threadIdx.x;
  long long total = (long long)E * C;
  if (tid >= total) return;
  int e = (int)(tid / C);
  int c = (int)(tid % C);
  int s = src[e];
  int d = dst[e];
  const float4* xp = (const float4*)(xsrc + (size_t)s * K);
  float4 v = xp[c];
  float* ap = agg + (size_t)d * K + (size_t)c * 4;
  atomicAdd(ap + 0, v.x);
  atomicAdd(ap + 1, v.y);
  atomicAdd(ap + 2, v.z);
  atomicAdd(ap + 3, v.w);
  if (c == 0) atomicAdd(cnt + d, 1.0f);
}

// ---------------- fused dual-GEMM + bias + L2 norm + relu ------------------
// Block: 256 threads = 8 waves. Block owns 16 rows x DOUT cols.
// Each wave computes DOUT/16/8 col-tiles of 16x16 via V_WMMA_F32_16X16X4_F32.
template <int K, int DOUT, bool RELU>
__global__ __launch_bounds__(256) void sage_gemm(
    const float* __restrict__ agg, const float* __restrict__ cnt,
    const float* __restrict__ xtgt, const float* __restrict__ Wl,
    const float* __restrict__ bl, const float* __restrict__ Wr,
    float* __restrict__ out) {
  __shared__ float ldsA[16 * K];  // agg rows, pre-divided by count
  __shared__ float ldsT[16 * K];  // target-node rows
  __shared__ float inv16[16];
  __shared__ float norm2[16];

  const int row0 = blockIdx.x * 16;
  const int tid = threadIdx.x;
  if (tid < 16) {
    inv16[tid] = 1.0f / fmaxf(cnt[row0 + tid], 1.0f);
    norm2[tid] = 0.0f;
  }
  __syncthreads();

  // Stage 16 x K tiles of both inputs into LDS (coalesced).
  #pragma unroll 2
  for (int idx = tid; idx < 16 * K; idx += 256) {
    int r = idx / K;
    ldsA[idx] = agg[(size_t)row0 * K + idx] * inv16[r];
    ldsT[idx] = xtgt[(size_t)row0 * K + idx];
  }
  __syncthreads();

  const int wave = tid >> 5;
  const int lane = tid & 31;
  const int nlo = lane & 15;  // tile column (B/C/D) and A row
  const int kh = lane >> 4;   // K-half selector for 16x16x4 layout

  constexpr int NT = DOUT / 16;       // column tiles in the block
  constexpr int TPW = (NT + 7) / 8;   // tiles per wave (1 or 2)
  v8f acc[TPW];

  #pragma unroll
  for (int t = 0; t < TPW; ++t) {
    const int col0 = (wave + t * 8) * 16;
    const float bias = bl[col0 + nlo];
    v8f c;
    #pragma unroll
    for (int r = 0; r < 8; ++r) c[r] = bias;  // bias folded into C

    // A(16x4 f32): lanes 0-15 -> K={0,1}, lanes 16-31 -> K={2,3}
    const float* Ap = ldsA + nlo * K + kh * 2;
    const float* Tp = ldsT + nlo * K + kh * 2;
    // B(4x16): B[k][n] = W[col0+n][k]
    const float* Blp = Wl + (size_t)(col0 + nlo) * K + kh * 2;
    const float* Brp = Wr + (size_t)(col0 + nlo) * K + kh * 2;

    #pragma unroll 4
    for (int k = 0; k < K; k += 4) {
      v2f a = *(const v2f*)(Ap + k);
      v2f b = *(const v2f*)(Blp + k);
      c = __builtin_amdgcn_wmma_f32_16x16x4_f32(false, a, false, b, (short)0,
                                                c, false, false);
    }
    #pragma unroll 4
    for (int k = 0; k < K; k += 4) {
      v2f a = *(const v2f*)(Tp + k);
      v2f b = *(const v2f*)(Brp + k);
      c = __builtin_amdgcn_wmma_f32_16x16x4_f32(false, a, false, b, (short)0,
                                                c, false, false);
    }
    acc[t] = c;
  }

  // Row-wise sum of squares: shuffle-reduce over 16 lanes, LDS-atomic across
  // waves/tiles. C layout: value (m = r + kh*8, n = nlo) in acc[t][r].
  #pragma unroll
  for (int t = 0; t < TPW; ++t) {
    #pragma unroll
    for (int r = 0; r < 8; ++r) {
      float v = acc[t][r];
      v *= v;
      v += __shfl_xor(v, 1, 16);
      v += __shfl_xor(v, 2, 16);
      v += __shfl_xor(v, 4, 16);
      v += __shfl_xor(v, 8, 16);
      if (nlo == 0) atomicAdd(&norm2[r + kh * 8], v);
    }
  }
  __syncthreads();

  // Normalize, ReLU, store.
  #pragma unroll
  for (int t = 0; t < TPW; ++t) {
    const int col0 = (wave + t * 8) * 16;
    #pragma unroll
    for (int r = 0; r < 8; ++r) {
      const int m = r + kh * 8;
      float nv = sqrtf(norm2[m]);
      float val = acc[t][r] / fmaxf(nv, SAGE_EPS);
      if (RELU) val = fmaxf(val, 0.0f);
      out[(size_t)(row0 + m) * DOUT + col0 + nlo] = val;
    }
  }
}

// ---------------------------------------------------------------------------
extern "C" void kernel_launch(void* const* d_in, const int* in_sizes, int n_in,
                              void* d_out, int out_size, void* d_ws,
                              size_t ws_size, hipStream_t stream) {
  const float* x   = (const float*)d_in[0];
  const int* src0  = (const int*)d_in[1];
  const int* dst0  = (const int*)d_in[2];
  const int* src1  = (const int*)d_in[3];
  const int* dst1  = (const int*)d_in[4];
  const int* src2  = (const int*)d_in[5];
  const int* dst2  = (const int*)d_in[6];
  const float* Wl0 = (const float*)d_in[7];
  const float* bl0 = (const float*)d_in[8];
  const float* Wr0 = (const float*)d_in[9];
  const float* Wl1 = (const float*)d_in[10];
  const float* bl1 = (const float*)d_in[11];
  const float* Wr1 = (const float*)d_in[12];
  const float* Wl2 = (const float*)d_in[13];
  const float* bl2 = (const float*)d_in[14];
  const float* Wr2 = (const float*)d_in[15];

  const int E0 = in_sizes[1], E1 = in_sizes[3], E2 = in_sizes[5];
  const int N1 = 294912, N2 = 49152, N3 = 8192;

  float* h1 = (float*)d_out;                 // [N1, 256]
  float* h2 = h1 + (size_t)N1 * 256;         // [N2, 256]
  float* h3 = h2 + (size_t)N2 * 256;         // [N3, 128]
  float* agg = (float*)d_ws;

  // ---- layer 0: 128 -> 256, relu ----
  {
    float* cnt = agg + (size_t)N1 * 128;
    long long n4 = ((long long)N1 * 128 + N1) / 4;
    sage_zero4<<<(unsigned)((n4 + 255) / 256), 256, 0, stream>>>((float4*)agg,
                                                                 n4);
    long long tot = (long long)E0 * (128 / 4);
    sage_scatter<128><<<(unsigned)((tot + 255) / 256), 256, 0, stream>>>(
        x, src0, dst0, agg, cnt, E0);
    sage_gemm<128, 256, true><<<N1 / 16, 256, 0, stream>>>(agg, cnt, x, Wl0,
                                                           bl0, Wr0, h1);
  }
  // ---- layer 1: 256 -> 256, relu ----
  {
    float* cnt = agg + (size_t)N2 * 256;
    long long n4 = ((long long)N2 * 256 + N2) / 4;
    sage_zero4<<<(unsigned)((n4 + 255) / 256), 256, 0, stream>>>((float4*)agg,
                                                                 n4);
    long long tot = (long long)E1 * (256 / 4);
    sage_scatter<256><<<(unsigned)((tot + 255) / 256), 256, 0, stream>>>(
        h1, src1, dst1, agg, cnt, E1);
    sage_gemm<256, 256, true><<<N2 / 16, 256, 0, stream>>>(agg, cnt, h1, Wl1,
                                                           bl1, Wr1, h2);
  }
  // ---- layer 2: 256 -> 128, no relu ----
  {
    float* cnt = agg + (size_t)N3 * 256;
    long long n4 = ((long long)N3 * 256 + N3) / 4;
    sage_zero4<<<(unsigned)((n4 + 255) / 256), 256, 0, stream>>>((float4*)agg,
                                                                 n4);
    long long tot = (long long)E2 * (256 / 4);
    sage_scatter<256><<<(unsigned)((tot + 255) / 256), 256, 0, stream>>>(
        h2, src2, dst2, agg, cnt, E2);
    sage_gemm<256, 128, false><<<N3 / 16, 256, 0, stream>>>(agg, cnt, h2, Wl2,
                                                            bl2, Wr2, h3);
  }
}